// NonGalerkinLoss2Eig_87041807221107
// MI455X (gfx1250) — compile-verified
//
#include <hip/hip_runtime.h>
#include <hip/hip_bf16.h>

// ---------------------------------------------------------------------------
// NonGalerkinLoss2Eig on MI455X (gfx1250, wave32, WMMA)
//
// Pipeline per dim d in {96,120,136,152,176}:
//   1) build_circ      : Phi circulant (zero-padded to np = ceil16(n))
//   2) gemm_wmma x5    : Phi^20 by binary exponentiation (FP32 WMMA 16x16x4).
//                        One WG per 16-wide C tile-column; 8 wave32s cover all
//                        tile rows. K-panels staged to LDS via
//                        GLOBAL_LOAD_ASYNC_TO_LDS_B32 (ASYNCcnt DMA),
//                        fragments fed from LDS.
//   3) jacobi_kernel   : S = tril(M)+tril(M,-1)^T built in LDS, parallel
//                        round-robin Jacobi eigensolver (S + Q fully LDS
//                        resident: 2*176^2*4B = 248KB < 320KB/WGP),
//                        ascending argsort, extract 25 eigenvector rows
//   4) loss_kernel     : AV = (PsiCirc - M20) contracted with V rows,
//                        sum of squares / K, global_atomic_add_f32
//
// Workspace layout (float): 4 regions of B*176*176 elements each
//   bufA | bufB | bufC | Vbuf   (~127 MB total -> L2 resident on MI455X)
// ---------------------------------------------------------------------------

typedef float v2f __attribute__((ext_vector_type(2)));
typedef float v8f __attribute__((ext_vector_type(8)));

typedef __attribute__((address_space(1))) int* gptr_i32;
typedef __attribute__((address_space(3))) int* lptr_i32;

#define NBATCH 65   // INPUT_SIZE
#define NTAPS  64   // INPUT_SIZE-1
#define KROWS  25

#if defined(__has_builtin)
#if __has_builtin(__builtin_amdgcn_global_load_async_to_lds_b32)
#define HAVE_ASYNC_LDS 1
#endif
#endif

// Stage one dword global -> LDS. Async (ASYNCcnt DMA, no VGPR data) when the
// toolchain exposes the gfx1250 builtin; plain copy otherwise.
__device__ __forceinline__ void stage_b32(const float* __restrict__ g, float* l) {
#ifdef HAVE_ASYNC_LDS
    __builtin_amdgcn_global_load_async_to_lds_b32(
        (gptr_i32)g, (lptr_i32)l, 0, 0);
#else
    *l = *g;
#endif
}

__device__ __forceinline__ void stage_wait() {
#ifdef HAVE_ASYNC_LDS
#if __has_builtin(__builtin_amdgcn_s_wait_asynccnt)
    __builtin_amdgcn_s_wait_asynccnt(0);
#else
    asm volatile("s_wait_asynccnt 0x0" ::: "memory");
#endif
#endif
    __syncthreads();
}

__global__ void zero_out_kernel(float* out) {
    if (blockIdx.x == 0 && threadIdx.x == 0) out[0] = 0.0f;
}

// Phi circulant, zero-padded to np x np. C[i,j] = phi_pad[(j-i) mod n].
__global__ void build_circ_kernel(const float* __restrict__ input,
                                  float* __restrict__ Phi,
                                  int n, int np, int p) {
    const int b = blockIdx.x;
    const float* __restrict__ phi = input + (size_t)b * NBATCH; // 64 taps + m col
    float* __restrict__ Pb = Phi + (size_t)b * np * np;
    for (int idx = threadIdx.x; idx < np * np; idx += blockDim.x) {
        const int i = idx / np;
        const int j = idx - i * np;
        float v = 0.0f;
        if (i < n && j < n) {
            int d = j - i;
            d += (d < 0) ? n : 0;
            if (d >= p && d < p + NTAPS) v = phi[d - p];
        }
        Pb[idx] = v;
    }
}

// Batched C = A * B, np x np (np % 16 == 0), FP32 WMMA 16x16x4.
// grid = (batch, np/16 tile-columns); block = 256 = 8 wave32.
// Wave w computes C tiles (w, tc) and (w+8, tc). Per 16-deep K stage the WG
// stages A[:, kk..kk+15] (np x 16) and B[kk..kk+15, tc-block] (16 x 16) into
// LDS with coalesced async-to-LDS copies; fragments then come from LDS.
// Tile-validity predicates are wave-uniform, so EXEC is all-ones at every
// WMMA (ISA 7.12) and every thread reaches every barrier.
__global__ void gemm_wmma_kernel(const float* __restrict__ A,
                                 const float* __restrict__ Bm,
                                 float* __restrict__ C, int np) {
    __shared__ float As[176 * 16];
    __shared__ float Bs[16 * 16];

    const int b = blockIdx.x;
    const int tc = blockIdx.y;             // tile column
    const int tiles_r = np >> 4;
    const int wave = threadIdx.x >> 5;     // 0..7
    const int lane = threadIdx.x & 31;
    const int half = lane >> 4;            // 0: K=0,1   1: K=2,3
    const int l = lane & 15;

    const size_t off = (size_t)b * np * np;
    const float* __restrict__ Ab = A + off;
    const float* __restrict__ Bb = Bm + off;
    float* __restrict__ Cb = C + off;

    const int bcol0 = tc << 4;
    const int bcol = bcol0 + l;

    const bool t0 = (wave < tiles_r);          // wave-uniform
    const bool t1 = (wave + 8 < tiles_r);      // wave-uniform
    v8f acc0 = {};
    v8f acc1 = {};

    for (int kk = 0; kk < np; kk += 16) {
        // ---- stage K panel (all threads participate) ----
        for (int e = threadIdx.x; e < np * 16; e += 256) {   // exact multiple
            const int row = e >> 4;
            const int c = e & 15;
            stage_b32(&Ab[(size_t)row * np + kk + c], &As[e]);
        }
        {
            const int r = threadIdx.x >> 4;
            const int c = threadIdx.x & 15;
            stage_b32(&Bb[(size_t)(kk + r) * np + bcol0 + c], &Bs[threadIdx.x]);
        }
        stage_wait();

        // ---- 4 WMMAs per owned tile from LDS ----
#pragma unroll
        for (int kc = 0; kc < 16; kc += 4) {
            const int ka = kc + (half << 1);
            v2f bf;
            bf.x = Bs[ka * 16 + l];
            bf.y = Bs[(ka + 1) * 16 + l];
            if (t0) {
                const int ar = (wave << 4) + l;
                v2f af;
                af.x = As[ar * 16 + ka];
                af.y = As[ar * 16 + ka + 1];
                acc0 = __builtin_amdgcn_wmma_f32_16x16x4_f32(
                    false, af, false, bf, (short)0, acc0, false, false);
            }
            if (t1) {
                const int ar = ((wave + 8) << 4) + l;
                v2f af;
                af.x = As[ar * 16 + ka];
                af.y = As[ar * 16 + ka + 1];
                acc1 = __builtin_amdgcn_wmma_f32_16x16x4_f32(
                    false, af, false, bf, (short)0, acc1, false, false);
            }
        }
        __syncthreads();   // LDS reused next stage
    }

    // C layout: VGPR r -> row r (lanes 0-15) / row r+8 (lanes 16-31)
    if (t0) {
        const int crow = (wave << 4) + (half << 3);
#pragma unroll
        for (int r = 0; r < 8; ++r)
            Cb[(size_t)(crow + r) * np + bcol] = acc0[r];
    }
    if (t1) {
        const int crow = ((wave + 8) << 4) + (half << 3);
#pragma unroll
        for (int r = 0; r < 8; ++r)
            Cb[(size_t)(crow + r) * np + bcol] = acc1[r];
    }
}

// One workgroup (256 threads = 8 wave32) per batch matrix.
// Builds S in LDS from M20, runs parallel-ordering (round-robin) two-sided
// Jacobi with accumulated rotations Q, argsorts eigenvalues ascending, and
// writes the 25 requested eigenvector-matrix rows (columns permuted) to V.
__global__ void jacobi_kernel(const float* __restrict__ M,
                              float* __restrict__ V,
                              int n, int np, int sweeps) {
    extern __shared__ float lds[];
    float* S  = lds;                 // np*np
    float* Q  = S + np * np;         // np*np
    float* cA = Q + np * np;         // up to 96
    float* sA = cA + 96;             // up to 96
    float* ev = sA + 96;             // up to 192
    int* perm = (int*)(ev + 192);    // up to 192
    int* top  = perm + 192;          // up to 96
    int* bot  = top + 96;            // up to 96

    const int b = blockIdx.x;
    const int tid = threadIdx.x;
    const int m = n >> 1;
    const float* __restrict__ Mb = M + (size_t)b * np * np;

    // S = tril(M) + tril(M,-1)^T  ;  Q = I
    for (int idx = tid; idx < n * n; idx += blockDim.x) {
        const int i = idx / n;
        const int j = idx - i * n;
        S[i * np + j] = (i >= j) ? Mb[i * np + j] : Mb[j * np + i];
        Q[i * np + j] = (i == j) ? 1.0f : 0.0f;
    }
    if (tid < m) { top[tid] = 2 * tid; bot[tid] = 2 * tid + 1; }
    __syncthreads();

    for (int sw = 0; sw < sweeps; ++sw) {
        for (int round = 0; round < n - 1; ++round) {
            // rotation angles for the m disjoint pairs
            if (tid < m) {
                int pp = top[tid], qq = bot[tid];
                if (pp > qq) { int t = pp; pp = qq; qq = t; }
                const float apq = S[pp * np + qq];
                const float app = S[pp * np + pp];
                const float aqq = S[qq * np + qq];
                float c = 1.0f, s = 0.0f;
                if (__builtin_fabsf(apq) > 1e-12f) {
                    const float tau = (aqq - app) / (2.0f * apq);
                    const float t = (tau >= 0.0f ? 1.0f : -1.0f) /
                                    (__builtin_fabsf(tau) + sqrtf(1.0f + tau * tau));
                    c = rsqrtf(1.0f + t * t);
                    s = t * c;
                }
                cA[tid] = c; sA[tid] = s;
            }
            __syncthreads();
            // left rotations: rows p,q
            for (int idx = tid; idx < m * n; idx += blockDim.x) {
                const int t = idx / n;
                const int j = idx - t * n;
                int pp = top[t], qq = bot[t];
                if (pp > qq) { int tt = pp; pp = qq; qq = tt; }
                const float c = cA[t], s = sA[t];
                const float sp = S[pp * np + j];
                const float sq = S[qq * np + j];
                S[pp * np + j] = c * sp - s * sq;
                S[qq * np + j] = s * sp + c * sq;
            }
            __syncthreads();
            // right rotations: cols p,q of S and Q
            for (int idx = tid; idx < m * n; idx += blockDim.x) {
                const int t = idx / n;
                const int i = idx - t * n;
                int pp = top[t], qq = bot[t];
                if (pp > qq) { int tt = pp; pp = qq; qq = tt; }
                const float c = cA[t], s = sA[t];
                float sp = S[i * np + pp];
                float sq = S[i * np + qq];
                S[i * np + pp] = c * sp - s * sq;
                S[i * np + qq] = s * sp + c * sq;
                sp = Q[i * np + pp];
                sq = Q[i * np + qq];
                Q[i * np + pp] = c * sp - s * sq;
                Q[i * np + qq] = s * sp + c * sq;
            }
            __syncthreads();
            // round-robin tournament rotation (player 0 fixed)
            if (tid == 0 && m > 1) {
                const int lastTop = top[m - 1];
                for (int i = m - 1; i >= 2; --i) top[i] = top[i - 1];
                const int b0 = bot[0];
                for (int i = 0; i < m - 1; ++i) bot[i] = bot[i + 1];
                bot[m - 1] = lastTop;
                top[1] = b0;
            }
            __syncthreads();
        }
    }

    // eigenvalues -> ascending argsort (matches eigh ordering)
    for (int i = tid; i < n; i += blockDim.x) ev[i] = S[i * np + i];
    __syncthreads();
    if (tid == 0) {
        for (int i = 0; i < n; ++i) perm[i] = i;
        for (int i = 0; i < n - 1; ++i) {
            int mi = i;
            for (int j = i + 1; j < n; ++j)
                if (ev[perm[j]] < ev[perm[mi]]) mi = j;
            const int t = perm[i]; perm[i] = perm[mi]; perm[mi] = t;
        }
    }
    __syncthreads();

    // V[k, j] = evecs_sorted[row_idx[k], j] = Q[row_idx[k], perm[j]]
    float* __restrict__ Vb = V + (size_t)b * 32 * np;
    for (int idx = tid; idx < KROWS * n; idx += blockDim.x) {
        const int k = idx / n;
        const int j = idx - k * n;
        const int r = (k == 0) ? 0 : (n - k);
        Vb[k * np + j] = Q[r * np + perm[j]];
    }
}

// loss += sum_k sum_i ( sum_j (PsiCirc(i,j) - M20[i,j]) * V[k,j] )^2 / K
__global__ void loss_kernel(const float* __restrict__ M,
                            const float* __restrict__ V,
                            const float* __restrict__ psi,
                            float* __restrict__ out,
                            int n, int np, int p) {
    __shared__ float psiS[NTAPS];
    __shared__ float vk[192];
    __shared__ float red[256];
    const int b = blockIdx.x;
    const int tid = threadIdx.x;
    const float* __restrict__ Mb = M + (size_t)b * np * np;
    const float* __restrict__ Vb = V + (size_t)b * 32 * np;
    if (tid < NTAPS) psiS[tid] = psi[b * NTAPS + tid];
    __syncthreads();

    float local = 0.0f;
    for (int k = 0; k < KROWS; ++k) {
        for (int j = tid; j < n; j += blockDim.x) vk[j] = Vb[k * np + j];
        __syncthreads();
        for (int i = tid; i < n; i += blockDim.x) {
            float dot = 0.0f;
            for (int j = 0; j < n; ++j) {
                int d = j - i;
                d += (d < 0) ? n : 0;
                float a = (d >= p && d < p + NTAPS) ? psiS[d - p] : 0.0f;
                a -= Mb[(size_t)i * np + j];
                dot = fmaf(a, vk[j], dot);
            }
            local = fmaf(dot, dot, local);
        }
        __syncthreads();
    }
    red[tid] = local;
    __syncthreads();
    for (int s = 128; s > 0; s >>= 1) {
        if (tid < s) red[tid] += red[tid + s];
        __syncthreads();
    }
    if (tid == 0) atomicAdd(out, red[0] / (float)KROWS);
}

extern "C" void kernel_launch(void* const* d_in, const int* in_sizes, int n_in,
                              void* d_out, int out_size, void* d_ws, size_t ws_size,
                              hipStream_t stream) {
    (void)in_sizes; (void)n_in; (void)out_size; (void)ws_size;
    const float* input   = (const float*)d_in[0];   // (256, 65) f32
    const float* outputs = (const float*)d_in[1];   // (256, 64) f32
    float* out = (float*)d_out;                     // scalar f32
    float* ws  = (float*)d_ws;

    const int B = 256;
    const size_t matElems = (size_t)B * 176 * 176;  // max np = 176
    float* bufA = ws;
    float* bufB = ws + matElems;
    float* bufC = ws + 2 * matElems;
    float* Vbuf = ws + 3 * matElems;

    zero_out_kernel<<<1, 32, 0, stream>>>(out);

    const int dims[5] = {96, 120, 136, 152, 176};
    for (int di = 0; di < 5; ++di) {
        const int n = dims[di];
        const int p = (n - NTAPS) / 2;
        const int np = (n + 15) & ~15;

        build_circ_kernel<<<B, 256, 0, stream>>>(input, bufA, n, np, p);

        dim3 g(B, np / 16);   // one WG per (batch, C tile-column)
        // Phi^20 = ((Phi^2)^2)^2^2 * Phi^4   (binary exponentiation)
        gemm_wmma_kernel<<<g, 256, 0, stream>>>(bufA, bufA, bufB, np); // M2
        gemm_wmma_kernel<<<g, 256, 0, stream>>>(bufB, bufB, bufC, np); // M4 (kept)
        gemm_wmma_kernel<<<g, 256, 0, stream>>>(bufC, bufC, bufA, np); // M8
        gemm_wmma_kernel<<<g, 256, 0, stream>>>(bufA, bufA, bufB, np); // M16
        gemm_wmma_kernel<<<g, 256, 0, stream>>>(bufB, bufC, bufA, np); // M20

        const size_t ldsBytes = (size_t)(2 * np * np + 1024) * sizeof(float);
        jacobi_kernel<<<B, 256, ldsBytes, stream>>>(bufA, Vbuf, n, np, /*sweeps=*/10);

        loss_kernel<<<B, 256, 0, stream>>>(bufA, Vbuf, outputs, out, n, np, p);
    }
}